// ColorUnpool_56650618634979
// MI455X (gfx1250) — compile-verified
//
#include <hip/hip_runtime.h>
#include <stdint.h>

// FEAT is fixed at 256 by the reference problem (row = 64 float4 = 1KB).
#define FEAT 256

// Get the 32-bit LDS byte offset of a __shared__ object (addrspace(3) ptrs are 32-bit).
typedef __attribute__((address_space(3))) char lds_char_t;
__device__ __forceinline__ unsigned lds_addr_of(void* p) {
  return (unsigned)(unsigned long long)(lds_char_t*)p;
}

// ---------------------------------------------------------------------------
// Kernel 1: copy feat rows to out[center_idx[row]] via the CDNA5 async
// global->LDS->global path. Each lane moves one 16B chunk; 256 threads/block
// move 4KB (4 rows) through a 4KB LDS staging buffer, never touching VGPRs
// for the payload. ASYNCcnt orders load-completion before the store reads
// LDS; S_ENDPGM's implicit wait-idle covers the final store.
// ---------------------------------------------------------------------------
__global__ void __launch_bounds__(256)
copy_centers_async(const float* __restrict__ feat,
                   const int* __restrict__ center_idx,
                   float* __restrict__ out,
                   long long n_chunks /* total float4 chunks = n_centers*64 */) {
  __shared__ __align__(16) float4 buf[256];
  long long i = (long long)blockIdx.x * 256 + threadIdx.x;
  if (i >= n_chunks) return;

  int row  = (int)(i >> 6);   // 64 float4 per 256-float row
  int col4 = (int)(i & 63);
  int drow = center_idx[row]; // arange in this problem, but honor it anyway

  const float4* src = (const float4*)feat + i;
  float4*       dst = (float4*)out + (long long)drow * (FEAT / 4) + col4;
  unsigned      lds = lds_addr_of(&buf[threadIdx.x]);

  asm volatile("global_load_async_to_lds_b128 %0, %1, off"
               :: "v"(lds), "v"(src) : "memory");
  asm volatile("s_wait_asynccnt 0" ::: "memory");
  asm volatile("global_store_async_from_lds_b128 %0, %1, off"
               :: "v"(dst), "v"(lds) : "memory");
}

// ---------------------------------------------------------------------------
// Kernel 2: zero the non-center rows with 128-bit stores (these must be
// exactly 0 before the atomic-max pass; also resets state between graph
// replays, making the whole launch sequence idempotent).
// ---------------------------------------------------------------------------
__global__ void __launch_bounds__(256)
zero_tail(float4* __restrict__ out4, long long n_chunks) {
  long long i = (long long)blockIdx.x * 256 + threadIdx.x;
  if (i < n_chunks) out4[i] = make_float4(0.f, 0.f, 0.f, 0.f);
}

// ---------------------------------------------------------------------------
// Kernel 3: one wave32 per edge. Lane L handles columns [8L, 8L+8):
// two coalesced b128 gathers from the (L2-resident) feat row, then eight
// non-returning GLOBAL_ATOMIC_MAX_NUM_F32 ops using the immediate-offset
// field. Starting from 0, atomic fmax == max(0, segment_max) exactly.
// ---------------------------------------------------------------------------
__global__ void __launch_bounds__(256)
scatter_edge_max(const float* __restrict__ feat,
                 const int* __restrict__ esrc,
                 const int* __restrict__ edst,
                 float* __restrict__ out,
                 int n_edges) {
  int gid  = blockIdx.x * 256 + threadIdx.x;
  int e    = gid >> 5;                 // wave32: one edge per wave
  int lane = threadIdx.x & 31;
  if (e >= n_edges) return;

  // Edge indices are wave-uniform; force them into SGPRs.
  int s = __builtin_amdgcn_readfirstlane(esrc[e]);
  int d = __builtin_amdgcn_readfirstlane(edst[e]);

  const float4* srow = (const float4*)(feat + (long long)s * FEAT);
  float4 a = srow[lane * 2 + 0];
  float4 b = srow[lane * 2 + 1];

  float* dp = out + (long long)d * FEAT + lane * 8;

#define AMAX(VAL, OFF)                                                  \
  asm volatile("global_atomic_max_num_f32 %0, %1, off offset:" OFF      \
               :: "v"(dp), "v"(VAL) : "memory")
  AMAX(a.x, "0");  AMAX(a.y, "4");  AMAX(a.z, "8");  AMAX(a.w, "12");
  AMAX(b.x, "16"); AMAX(b.y, "20"); AMAX(b.z, "24"); AMAX(b.w, "28");
#undef AMAX
}

// ---------------------------------------------------------------------------
// Host launcher. Inputs (setup_inputs order):
//   d_in[0] feat       float32 [n_centers, FEAT]
//   d_in[1] center_idx int32   [n_centers]
//   d_in[2] edge_src   int32   [n_edges]
//   d_in[3] edge_dst   int32   [n_edges]
//   d_in[4] n_nodes    int32   scalar (derived from out_size instead)
// ---------------------------------------------------------------------------
extern "C" void kernel_launch(void* const* d_in, const int* in_sizes, int n_in,
                              void* d_out, int out_size, void* d_ws, size_t ws_size,
                              hipStream_t stream) {
  const float* feat       = (const float*)d_in[0];
  const int*   center_idx = (const int*)d_in[1];
  const int*   esrc       = (const int*)d_in[2];
  const int*   edst       = (const int*)d_in[3];
  float*       out        = (float*)d_out;

  const int n_centers = in_sizes[1];
  const int n_edges   = in_sizes[2];
  const int feat_dim  = in_sizes[0] / n_centers;          // == 256
  const long long n_nodes = (long long)out_size / feat_dim;

  // 1) centers: async LDS-staged copy of feat into out rows [0, n_centers)
  const long long c4 = (long long)n_centers * feat_dim / 4;
  copy_centers_async<<<(unsigned)((c4 + 255) / 256), 256, 0, stream>>>(
      feat, center_idx, out, c4);

  // 2) zero rows [n_centers, n_nodes)
  const long long z4 = (n_nodes - n_centers) * feat_dim / 4;
  zero_tail<<<(unsigned)((z4 + 255) / 256), 256, 0, stream>>>(
      (float4*)out + (long long)n_centers * (feat_dim / 4), z4);

  // 3) atomic fmax scatter, one wave per edge (runs after zero via stream order)
  const long long sthreads = (long long)n_edges * 32;
  scatter_edge_max<<<(unsigned)((sthreads + 255) / 256), 256, 0, stream>>>(
      feat, esrc, edst, out, n_edges);
}